// RGA_28475633173113
// MI455X (gfx1250) — compile-verified
//
#include <hip/hip_runtime.h>

typedef __attribute__((ext_vector_type(2))) float v2f;
typedef __attribute__((ext_vector_type(8))) float v8f;

#define B_ 4
#define N_ 32768
#define EPS_ 1e-6f
#define LRELU_ 0.2f

// ---------------------------------------------------------------------------
// Fused GEMM:  Out[b,i,:] = act( (A[b, gidx?idx[b,i]:i, :] @ W + Cadd + bias) * bn_s + bn_t )
//   A        : (B, *, Cin) with batch stride strideA (rows may exceed n; only n used)
//   gidx     : optional (B, n) row gather indices into A
//   W        : (Cin, Cout) row-major
//   Cadd     : optional (B, n, Cout) pre-accumulator (for split-concat decoder GEMMs)
//   act      : 0 = none, 1 = relu, 2 = leaky(0.2)
// One wave = one 16x(16*NT) fp32 tile via V_WMMA_F32_16X16X4_F32; the A fragment
// is loaded once per K-step and reused across NT wmmas (A loads are the only
// uncoalesced accesses, so NT-blocking amortizes them NTx).
// ---------------------------------------------------------------------------
template <int NT>
__global__ void gemm16_wmma(const float* __restrict__ A, long long strideA, int Cin,
                            const int* __restrict__ gidx,
                            const float* __restrict__ W,
                            const float* __restrict__ bias,
                            const float* __restrict__ gamma,
                            const float* __restrict__ beta,
                            const float* __restrict__ Cadd,
                            float* __restrict__ Out,
                            int n, int Cout, int tilesM, int tilesN, int act,
                            long long totalTiles) {
    int wave = threadIdx.x >> 5;
    int lane = threadIdx.x & 31;
    long long tile = (long long)blockIdx.x * (blockDim.x >> 5) + wave;
    if (tile >= totalTiles) return;                 // wave-uniform exit

    long long tilesPerBatch = (long long)tilesM * tilesN;
    int b  = (int)(tile / tilesPerBatch);
    int t  = (int)(tile % tilesPerBatch);
    int tm = t / tilesN;
    int tn = t % tilesN;                            // coarse N tile (16*NT cols)

    int half = lane >> 4;      // 0: lanes 0-15, 1: lanes 16-31
    int l16  = lane & 15;
    int col0 = tn * (16 * NT) + l16;

    // A row owned by this lane for the A-fragment (M = l16)
    int rowIdx = tm * 16 + l16;
    long long aRow = gidx ? (long long)gidx[(long long)b * n + rowIdx] : (long long)rowIdx;
    const float* Ar = A + (long long)b * strideA + aRow * (long long)Cin;
    const float* Wc = W + col0;

    v8f acc[NT];
    #pragma unroll
    for (int nt = 0; nt < NT; nt++) acc[nt] = (v8f){0.f, 0.f, 0.f, 0.f, 0.f, 0.f, 0.f, 0.f};

    int koff = 2 * half;       // lanes 16-31 carry K=+2,+3
    #pragma unroll 2
    for (int k = 0; k < Cin; k += 4) {
        v2f a = *(const v2f*)(Ar + k + koff);           // A[M=l16, K=k+koff .. +1]
        const float* W0 = Wc + (k + koff) * Cout;       // row K=k+koff
        const float* W1 = W0 + Cout;                    // row K=k+koff+1
        #pragma unroll
        for (int nt = 0; nt < NT; nt++) {
            v2f bfrag;
            bfrag.x = W0[nt * 16];
            bfrag.y = W1[nt * 16];
            acc[nt] = __builtin_amdgcn_wmma_f32_16x16x4_f32(
                /*neg_a=*/false, a, /*neg_b=*/false, bfrag,
                /*c_mod=*/(short)0, acc[nt], /*reuse_a=*/false, /*reuse_b=*/false);
        }
    }

    long long outBase = (long long)b * n * Cout;
    #pragma unroll
    for (int nt = 0; nt < NT; nt++) {
        int col = col0 + nt * 16;
        float bsv = bias ? bias[col] : 0.f;
        float s = 1.f, bt = 0.f;
        if (gamma) { s = gamma[col] * rsqrtf(1.f + EPS_); bt = beta[col]; }
        #pragma unroll
        for (int v = 0; v < 8; v++) {
            int row = tm * 16 + v + 8 * half;           // C/D layout: halves offset rows by 8
            long long o = outBase + (long long)row * Cout + col;
            float val = acc[nt][v];
            if (Cadd)  val += Cadd[o];
            val += bsv;
            if (gamma) val = val * s + bt;
            if (act == 1)      val = val > 0.f ? val : 0.f;
            else if (act == 2) val = val > 0.f ? val : LRELU_ * val;
            Out[o] = val;
        }
    }
}

// ---------------------------------------------------------------------------
// 1-NN: for each query q in coords[b, 0..Q) find argmin over support coords[b, 0..S)
// of  |q|^2 - 2 q.s + |s|^2  (reference's expanded form; first-min wins).
// ---------------------------------------------------------------------------
#define KTILE 128
__global__ void knn1_kernel(const float* __restrict__ coords, int Q, int S,
                            long long strideB, int* __restrict__ idx) {
    __shared__ float4 tile[KTILE];
    int qBlocks = Q >> 8;                            // Q / 256
    int b = blockIdx.x / qBlocks;
    int q = (blockIdx.x % qBlocks) * 256 + threadIdx.x;
    const float* cb = coords + (long long)b * strideB;

    float qx = cb[q * 3 + 0], qy = cb[q * 3 + 1], qz = cb[q * 3 + 2];
    float q2 = qx * qx + qy * qy + qz * qz;
    float best = 3.4e38f;
    int bi = 0;
    for (int s0 = 0; s0 < S; s0 += KTILE) {
        if (threadIdx.x < KTILE) {
            int si = s0 + threadIdx.x;
            float sx = cb[si * 3 + 0], sy = cb[si * 3 + 1], sz = cb[si * 3 + 2];
            tile[threadIdx.x] = make_float4(sx, sy, sz, sx * sx + sy * sy + sz * sz);
        }
        __syncthreads();
        #pragma unroll 8
        for (int j = 0; j < KTILE; j++) {
            float4 sp = tile[j];
            float d = q2 - 2.f * (qx * sp.x + qy * sp.y + qz * sp.z) + sp.w;
            if (d < best) { best = d; bi = s0 + j; }
        }
        __syncthreads();
    }
    idx[(long long)b * Q + q] = bi;
}

// ---------------------------------------------------------------------------
// Start layer: x = leaky(bn(input @ W6x64 + b)), with permutation gather fused,
// also emits permuted coords = input[..., :3][:, perm].
// ---------------------------------------------------------------------------
__global__ void start_kernel(const float* __restrict__ inp, const int* __restrict__ perm,
                             const float* __restrict__ W, const float* __restrict__ bias,
                             const float* __restrict__ g, const float* __restrict__ bt,
                             float* __restrict__ xp, float* __restrict__ coords) {
    __shared__ float sW[6 * 64];
    __shared__ float sB[64], sS[64], sT[64];
    for (int t = threadIdx.x; t < 384; t += blockDim.x) sW[t] = W[t];
    if (threadIdx.x < 64) {
        sB[threadIdx.x] = bias[threadIdx.x];
        sS[threadIdx.x] = g[threadIdx.x] * rsqrtf(1.f + EPS_);
        sT[threadIdx.x] = bt[threadIdx.x];
    }
    __syncthreads();
    long long gid = (long long)blockIdx.x * blockDim.x + threadIdx.x;   // over B*N
    int b = (int)(gid / N_);
    int i = (int)(gid % N_);
    int j = perm[i];
    const float* r = inp + ((long long)b * N_ + j) * 6;
    float v0 = r[0], v1 = r[1], v2 = r[2], v3 = r[3], v4 = r[4], v5 = r[5];
    float* cd = coords + ((long long)b * N_ + i) * 3;
    cd[0] = v0; cd[1] = v1; cd[2] = v2;
    float* xo = xp + ((long long)b * N_ + i) * 64;
    #pragma unroll 8
    for (int c = 0; c < 64; c++) {
        float a = sB[c] + v0 * sW[c] + v1 * sW[64 + c] + v2 * sW[128 + c]
                + v3 * sW[192 + c] + v4 * sW[256 + c] + v5 * sW[320 + c];
        a = a * sS[c] + sT[c];
        xo[c] = a > 0.f ? a : LRELU_ * a;
    }
}

// Inverse permutation scatter: out[b, perm[i], :] = in[b, i, :]  (C = 32)
__global__ void unperm_kernel(const float* __restrict__ in, const int* __restrict__ perm,
                              float* __restrict__ out) {
    long long gid = (long long)blockIdx.x * blockDim.x + threadIdx.x;   // over B*N
    int b = (int)(gid / N_);
    int i = (int)(gid % N_);
    int j = perm[i];
    const float4* src = (const float4*)(in + ((long long)b * N_ + i) * 32);
    float4* dst = (float4*)(out + ((long long)b * N_ + j) * 32);
    #pragma unroll
    for (int k = 0; k < 8; k++) dst[k] = src[k];
}

// Final layer 32 -> 13 with fused transpose: out[b, c, n] = (x[b,n,:] @ W)[c] + bias[c]
__global__ void final_kernel(const float* __restrict__ x, const float* __restrict__ W,
                             const float* __restrict__ bias, float* __restrict__ out) {
    __shared__ float sW[32 * 13];
    __shared__ float sB[13];
    for (int t = threadIdx.x; t < 416; t += blockDim.x) sW[t] = W[t];
    if (threadIdx.x < 13) sB[threadIdx.x] = bias[threadIdx.x];
    __syncthreads();
    long long gid = (long long)blockIdx.x * blockDim.x + threadIdx.x;   // over B*N
    int b = (int)(gid / N_);
    int i = (int)(gid % N_);
    const float* r = x + ((long long)b * N_ + i) * 32;
    float v[32];
    #pragma unroll
    for (int k = 0; k < 32; k++) v[k] = r[k];
    #pragma unroll
    for (int c = 0; c < 13; c++) {
        float a = sB[c];
        #pragma unroll
        for (int k = 0; k < 32; k++) a += v[k] * sW[k * 13 + c];
        out[((long long)b * 13 + c) * N_ + i] = a;
    }
}

// ---------------------------------------------------------------------------
// Workspace arena (floats). Total = 55,050,240 floats (~220 MB).
// ---------------------------------------------------------------------------
static const long long OFF_COORDS = 0;                        // B*N*3    =   393216
static const long long OFF_IDX    = 393216;                   // B*N ints =   131072
static const long long OFF_X0     = 524288;                   // B*N*64   =  8388608 (reused for end1 out)
static const long long OFF_E1     = OFF_X0 + 8388608;         //  8388608
static const long long OFF_E2     = OFF_E1 + 8388608;         //  8388608
static const long long OFF_E3     = OFF_E2 + 8388608;         //  8388608
static const long long OFF_E4     = OFF_E3 + 8388608;         //  8388608
static const long long OFF_BUFA   = OFF_E4 + 8388608;         //  4194304 (mlp / dec outs / end2 out)
static const long long OFF_BUFT   = OFF_BUFA + 4194304;       //  4194304 (decoder pass-1 accumulator)
static const long long OFF_BUFU   = OFF_BUFT + 4194304;       //  4194304 (un-permuted x)

extern "C" void kernel_launch(void* const* d_in, const int* in_sizes, int n_in,
                              void* d_out, int out_size, void* d_ws, size_t ws_size,
                              hipStream_t stream) {
    const float* input = (const float*)d_in[0];
    const int*   perm  = (const int*)d_in[1];
    // p_start: W,b,g,bt -> d_in[2..5]
    // p_enc blocks: base 6, 4 each; p_mlp: 22,23; p_dec blocks: base 24, 4 each; p_end: 40..49
    const float* in_f[50];
    for (int i = 2; i < 50; i++) in_f[i] = (const float*)d_in[i];

    float* ws     = (float*)d_ws;
    float* coords = ws + OFF_COORDS;
    int*   idx    = (int*)(ws + OFF_IDX);
    float* x0     = ws + OFF_X0;
    float* e1     = ws + OFF_E1;
    float* e2     = ws + OFF_E2;
    float* e3     = ws + OFF_E3;
    float* e4     = ws + OFF_E4;
    float* bufA   = ws + OFF_BUFA;
    float* bufT   = ws + OFF_BUFT;
    float* bufU   = ws + OFF_BUFU;
    float* out    = (float*)d_out;

    auto gemm = [&](const float* A, long long strideA, int Cin, const int* gi,
                    const float* W, const float* bias, const float* g, const float* bt,
                    const float* Cadd, float* Out, int n, int Cout, int act) {
        int tilesM = n / 16;
        if (Cout % 64 == 0) {
            int tilesN = Cout / 64;                     // NT = 4 -> 16x64 per wave
            long long total = (long long)B_ * tilesM * tilesN;
            int blocks = (int)((total + 7) / 8);        // 8 waves (256 threads) per block
            hipLaunchKernelGGL((gemm16_wmma<4>), dim3(blocks), dim3(256), 0, stream,
                               A, strideA, Cin, gi, W, bias, g, bt, Cadd, Out,
                               n, Cout, tilesM, tilesN, act, total);
        } else {
            int tilesN = Cout / 32;                     // NT = 2 -> 16x32 per wave
            long long total = (long long)B_ * tilesM * tilesN;
            int blocks = (int)((total + 7) / 8);
            hipLaunchKernelGGL((gemm16_wmma<2>), dim3(blocks), dim3(256), 0, stream,
                               A, strideA, Cin, gi, W, bias, g, bt, Cadd, Out,
                               n, Cout, tilesM, tilesN, act, total);
        }
    };
    auto knn = [&](int Q, int S) {
        hipLaunchKernelGGL(knn1_kernel, dim3(B_ * (Q / 256)), dim3(256), 0, stream,
                           coords, Q, S, (long long)N_ * 3, idx);
    };

    // ---- start (6 -> 64, BN + leaky, fused permutation gather) ----
    hipLaunchKernelGGL(start_kernel, dim3(B_ * N_ / 256), dim3(256), 0, stream,
                       input, perm, in_f[2], in_f[3], in_f[4], in_f[5], x0, coords);

    // ---- encoder (leaky + BN), stack = e1..e4 ----
    gemm(x0, (long long)N_ * 64,    64,  nullptr, in_f[6],  in_f[7],  in_f[8],  in_f[9],  nullptr, e1, 32768, 64,  2);
    gemm(e1, (long long)32768 * 64, 64,  nullptr, in_f[10], in_f[11], in_f[12], in_f[13], nullptr, e2, 16384, 128, 2);
    gemm(e2, (long long)16384 * 128,128, nullptr, in_f[14], in_f[15], in_f[16], in_f[17], nullptr, e3, 8192,  256, 2);
    gemm(e3, (long long)8192 * 256, 256, nullptr, in_f[18], in_f[19], in_f[20], in_f[21], nullptr, e4, 4096,  512, 2);

    // ---- bottleneck MLP (relu, no BN) ----
    gemm(e4, (long long)4096 * 512, 512, nullptr, in_f[22], in_f[23], nullptr, nullptr, nullptr, bufA, 2048, 512, 1);

    // ---- decoder: concat([gather(x, knn), skip]) @ W  ==  gather(x)@W_top + skip@W_bot ----
    // level 0: fine 4096, support 2048, 1024 -> 256
    knn(4096, 2048);
    gemm(bufA, (long long)2048 * 512, 512, idx,     in_f[24],             nullptr,  nullptr,  nullptr,  nullptr, bufT, 4096, 256, 0);
    gemm(e4,   (long long)4096 * 512, 512, nullptr, in_f[24] + 512 * 256, in_f[25], in_f[26], in_f[27], bufT,    bufA, 4096, 256, 1);
    // level 1: fine 8192, support 4096, 512 -> 128
    knn(8192, 4096);
    gemm(bufA, (long long)4096 * 256, 256, idx,     in_f[28],             nullptr,  nullptr,  nullptr,  nullptr, bufT, 8192, 128, 0);
    gemm(e3,   (long long)8192 * 256, 256, nullptr, in_f[28] + 256 * 128, in_f[29], in_f[30], in_f[31], bufT,    bufA, 8192, 128, 1);
    // level 2: fine 16384, support 8192, 256 -> 64
    knn(16384, 8192);
    gemm(bufA, (long long)8192 * 128, 128, idx,     in_f[32],             nullptr,  nullptr,  nullptr,  nullptr, bufT, 16384, 64, 0);
    gemm(e2,   (long long)16384 * 128,128, nullptr, in_f[32] + 128 * 64,  in_f[33], in_f[34], in_f[35], bufT,    bufA, 16384, 64, 1);
    // level 3: fine 32768, support 16384, 128 -> 32
    knn(32768, 16384);
    gemm(bufA, (long long)16384 * 64, 64,  idx,     in_f[36],             nullptr,  nullptr,  nullptr,  nullptr, bufT, 32768, 32, 0);
    gemm(e1,   (long long)32768 * 64, 64,  nullptr, in_f[36] + 64 * 32,   in_f[37], in_f[38], in_f[39], bufT,    bufA, 32768, 32, 1);

    // ---- undo permutation ----
    hipLaunchKernelGGL(unperm_kernel, dim3(B_ * N_ / 256), dim3(256), 0, stream, bufA, perm, bufU);

    // ---- head: 32 -> 64 -> 32 (relu + BN), then 32 -> 13 with transpose ----
    gemm(bufU, (long long)N_ * 32, 32, nullptr, in_f[40], in_f[41], in_f[42], in_f[43], nullptr, x0,   32768, 64, 1);
    gemm(x0,   (long long)N_ * 64, 64, nullptr, in_f[44], in_f[45], in_f[46], in_f[47], nullptr, bufA, 32768, 32, 1);
    hipLaunchKernelGGL(final_kernel, dim3(B_ * N_ / 256), dim3(256), 0, stream,
                       bufA, in_f[48], in_f[49], out);
}